// Gate_29738353558213
// MI455X (gfx1250) — compile-verified
//
#include <hip/hip_runtime.h>
#include <hip/hip_bf16.h>

typedef __attribute__((ext_vector_type(2))) float v2f;
typedef __attribute__((ext_vector_type(8))) float v8f;

// async-LDS builtin pointer types (per hipcc diagnostic: int __vector(N) __device__ *)
typedef int vi4 __attribute__((vector_size(16)));
typedef int vi2 __attribute__((vector_size(8)));
#define GAS __attribute__((address_space(1)))
#define LAS __attribute__((address_space(3)))

#define N_TOKENS 16384
#define MODEL_DIM 2048
#define NUM_EXPERTS 16
#define CAPACITY 2048
#define NK (N_TOKENS * 2)        // 32768 routed entries
#define TOK_PER_BLOCK 128
#define NUM_BLOCKS (N_TOKENS / TOK_PER_BLOCK)   // 128
#define KC 32                    // K chunk through LDS
#define NCHUNK (MODEL_DIM / KC)  // 64
#define XPAD 36                  // padded row stride (floats): 144B = 16B-aligned, conflict-free banks

#if __has_builtin(__builtin_amdgcn_global_load_async_to_lds_b128) && \
    __has_builtin(__builtin_amdgcn_global_load_async_to_lds_b64)
#define HAVE_ASYNC_LDS 1
#else
#define HAVE_ASYNC_LDS 0
#endif

__device__ __forceinline__ void wait_async0() {
#if __has_builtin(__builtin_amdgcn_s_wait_asynccnt)
    __builtin_amdgcn_s_wait_asynccnt(0);
#else
    asm volatile("s_wait_asynccnt 0" ::: "memory");
#endif
}

// ---------- 16-lane-group reductions (wave32: xor masks <=8 stay in each half) ----------
__device__ __forceinline__ float grp_max16(float v) {
    for (int m = 8; m >= 1; m >>= 1) v = fmaxf(v, __shfl_xor(v, m, 32));
    return v;
}
__device__ __forceinline__ float grp_sum16(float v) {
    for (int m = 8; m >= 1; m >>= 1) v += __shfl_xor(v, m, 32);
    return v;
}
// argmax over 16 lanes, ties -> lowest index (matches jnp.argmax / lax.top_k)
__device__ __forceinline__ void grp_argmax16(float v, int i, float& mv, int& mi) {
    mv = v; mi = i;
    for (int m = 8; m >= 1; m >>= 1) {
        float ov = __shfl_xor(mv, m, 32);
        int   oi = __shfl_xor(mi, m, 32);
        if (ov > mv || (ov == mv && oi < mi)) { mv = ov; mi = oi; }
    }
}

// ================= Kernel 1: fused gate GEMM (WMMA f32 16x16x4) + softmax/top2/partials =================
__global__ __launch_bounds__(256)
void Gate_29738353558213_gate_kernel(const float* __restrict__ x,
                                     const float* __restrict__ wg,
                                     float* __restrict__ outW,    // [N,2] top-2 gate values
                                     int*   __restrict__ topE,    // [N*2] top-2 expert ids (scratch)
                                     float* __restrict__ meP,     // [NUM_BLOCKS*16] partial sum of gates
                                     float* __restrict__ ceP)     // [NUM_BLOCKS*16] partial argmax counts
{
#if HAVE_ASYNC_LDS
    __shared__ float sX[2][TOK_PER_BLOCK * XPAD];   // double-buffered 128x32 x tile
    __shared__ float sW[2][NUM_EXPERTS * XPAD];     // double-buffered 16x32 wg tile
#else
    __shared__ float sX[1][TOK_PER_BLOCK * XPAD];
    __shared__ float sW[1][NUM_EXPERTS * XPAD];
#endif
    __shared__ float sMeP[8][32];
    __shared__ int   sArg[TOK_PER_BLOCK];

    const int tid  = threadIdx.x;
    const int wave = tid >> 5;
    const int lane = tid & 31;
    const int tokBlock = blockIdx.x * TOK_PER_BLOCK;

    const int m    = lane & 15;         // A row within tile / B column (expert)
    const int koff = (lane >> 4) * 2;   // K sub-offset per ISA A/B f32 layout

    v8f c = {0.f, 0.f, 0.f, 0.f, 0.f, 0.f, 0.f, 0.f};

    // uniform loader mappings (all 256 lanes active -> no exec-mask branches)
    const int xcol  = (tid & 7) * 4;    // X: 8 lanes cover one 32-float row (128B contiguous)
    const int xrow0 = tid >> 3;         // 0..31, stride 32 over 128 rows
    const int wrow  = tid >> 4;         // W: 16 lanes cover one row via b64
    const int wcol  = (tid & 15) * 2;

    const float* xg = x + (size_t)tokBlock * MODEL_DIM;

#if HAVE_ASYNC_LDS
    // ---- async global->LDS DMA, double-buffered: prefetch(i+1) overlaps compute(i) ----
    auto prefetch = [&](int k0, int buf) {
        #pragma unroll
        for (int p = 0; p < 4; ++p) {
            const int r = xrow0 + p * 32;
            __builtin_amdgcn_global_load_async_to_lds_b128(
                (GAS vi4*)(xg + (size_t)r * MODEL_DIM + k0 + xcol),
                (LAS vi4*)&sX[buf][r * XPAD + xcol], 0, 0);
        }
        __builtin_amdgcn_global_load_async_to_lds_b64(
            (GAS vi2*)(wg + (size_t)wrow * MODEL_DIM + k0 + wcol),
            (LAS vi2*)&sW[buf][wrow * XPAD + wcol], 0, 0);
    };

    prefetch(0, 0);
    for (int i = 0; i < NCHUNK; ++i) {
        const int buf = i & 1;
        wait_async0();          // own chunk-i writes landed
        __syncthreads();        // everyone's landed; everyone done reading buf^1
        if (i + 1 < NCHUNK) prefetch((i + 1) * KC, buf ^ 1);

        const float* bx = &sX[buf][(wave * 16 + m) * XPAD];
        const float* bw = &sW[buf][m * XPAD];
        #pragma unroll
        for (int kk = 0; kk < KC; kk += 4) {
            v2f a = *(const v2f*)(bx + kk + koff);  // A[m][koff..koff+1]
            v2f b = *(const v2f*)(bw + kk + koff);  // B[koff..][n=m] = wg[n][k]
            c = __builtin_amdgcn_wmma_f32_16x16x4_f32(
                    false, a, false, b, (short)0, c, false, false);
        }
    }
#else
    // ---- fallback: batch all global loads to VGPRs, then store to LDS ----
    for (int i = 0; i < NCHUNK; ++i) {
        const int k0 = i * KC;
        float4 t[4];
        #pragma unroll
        for (int p = 0; p < 4; ++p)
            t[p] = *(const float4*)(xg + (size_t)(xrow0 + p * 32) * MODEL_DIM + k0 + xcol);
        float2 tw = *(const float2*)(wg + (size_t)wrow * MODEL_DIM + k0 + wcol);
        __syncthreads();
        #pragma unroll
        for (int p = 0; p < 4; ++p)
            *(float4*)&sX[0][(xrow0 + p * 32) * XPAD + xcol] = t[p];
        *(float2*)&sW[0][wrow * XPAD + wcol] = tw;
        __syncthreads();

        const float* bx = &sX[0][(wave * 16 + m) * XPAD];
        const float* bw = &sW[0][m * XPAD];
        #pragma unroll
        for (int kk = 0; kk < KC; kk += 4) {
            v2f a = *(const v2f*)(bx + kk + koff);
            v2f b = *(const v2f*)(bw + kk + koff);
            c = __builtin_amdgcn_wmma_f32_16x16x4_f32(
                    false, a, false, b, (short)0, c, false, false);
        }
    }
#endif

    // ---- epilogue: C VGPR j holds token M=j (lanes 0-15) / M=j+8 (lanes 16-31), N = lane&15 ----
    const int e    = m;
    const int half = lane >> 4;
    float me_acc = 0.f;

    #pragma unroll
    for (int j = 0; j < 8; ++j) {
        const float logit = c[j];
        const float mx = grp_max16(logit);
        const float p  = __expf(logit - mx);
        const float s  = grp_sum16(p);
        const float gate = p / s;
        me_acc += gate;

        float v1; int i1;
        grp_argmax16(gate, e, v1, i1);             // top-1 (also argmax for ce)
        const float g2 = (e == i1) ? -1.0f : gate; // gates > 0, so -1 excludes top-1
        float v2; int i2;
        grp_argmax16(g2, e, v2, i2);               // top-2

        const int tok = tokBlock + wave * 16 + half * 8 + j;
        if (e == 0) {
            outW[tok * 2 + 0] = v1;
            outW[tok * 2 + 1] = v2;
            topE[tok * 2 + 0] = i1;
            topE[tok * 2 + 1] = i2;
            sArg[wave * 16 + half * 8 + j] = i1;
        }
    }
    sMeP[wave][lane] = me_acc;
    __syncthreads();

    // deterministic fixed-order block reduction (no float atomics -> replay-stable)
    if (tid < 16) {
        float ms = 0.f;
        #pragma unroll
        for (int w = 0; w < 8; ++w) ms += sMeP[w][tid] + sMeP[w][tid + 16];
        meP[blockIdx.x * 16 + tid] = ms;
        float cnt = 0.f;
        for (int t = 0; t < TOK_PER_BLOCK; ++t) cnt += (sArg[t] == tid) ? 1.0f : 0.0f;
        ceP[blockIdx.x * 16 + tid] = cnt;
    }
}

// ================= Kernel 2: stable expert-major dispatch (one block per expert) =================
__global__ __launch_bounds__(256)
void Gate_29738353558213_dispatch_kernel(const int* __restrict__ topE,
                                         float* __restrict__ outIdx,   // [E*CAPACITY], -1 = empty
                                         float* __restrict__ outEid)   // [E*CAPACITY]
{
    const int e = blockIdx.x;
    for (int i = threadIdx.x; i < CAPACITY; i += blockDim.x) {
        outIdx[e * CAPACITY + i] = -1.0f;
        outEid[e * CAPACITY + i] = (float)e;
    }
    __syncthreads();

    if (threadIdx.x < 32) {
        const int lane = threadIdx.x;
        int base = 0;   // running count of entries for this expert (uniform across lanes)
        for (int i0 = 0; i0 < NK; i0 += 32) {
            const int ex = topE[i0 + lane];
            const bool match = (ex == e);
            const unsigned int b32 = (unsigned int)__ballot(match);
            const int prefix = __popc(b32 & ((1u << lane) - 1u));
            if (match) {
                const int pos = base + prefix;                 // stable rank (ascending flat idx)
                if (pos < CAPACITY)
                    outIdx[e * CAPACITY + pos] = (float)((i0 + lane) >> 1);  // token id
            }
            base += __popc(b32);
        }
    }
}

// ================= Kernel 3: l_aux from partials, fixed summation order =================
__global__ void Gate_29738353558213_laux_kernel(const float* __restrict__ meP,
                                                const float* __restrict__ ceP,
                                                float* __restrict__ out)
{
    if (threadIdx.x == 0 && blockIdx.x == 0) {
        float l = 0.f;
        for (int e = 0; e < NUM_EXPERTS; ++e) {
            float me = 0.f, ce = 0.f;
            for (int b = 0; b < NUM_BLOCKS; ++b) {
                me += meP[b * 16 + e];
                ce += ceP[b * 16 + e];
            }
            l += (me * (1.0f / N_TOKENS)) * (ce * (1.0f / N_TOKENS));
        }
        out[0] = l * (float)NUM_EXPERTS;
    }
}

extern "C" void kernel_launch(void* const* d_in, const int* in_sizes, int n_in,
                              void* d_out, int out_size, void* d_ws, size_t ws_size,
                              hipStream_t stream) {
    const float* x  = (const float*)d_in[0];   // [16384, 2048]
    const float* wg = (const float*)d_in[1];   // [16, 2048]

    float* out = (float*)d_out;
    float* out_laux = out;                       // [1]
    float* out_w    = out + 1;                   // [N*2]
    float* out_idx  = out + 1 + NK;              // [E*CAP]
    float* out_eid  = out + 1 + 2 * NK;          // [E*CAP]

    float* ws   = (float*)d_ws;
    float* meP  = ws;                            // [128*16]
    float* ceP  = ws + NUM_BLOCKS * 16;          // [128*16]
    int*   topE = (int*)(ws + 2 * NUM_BLOCKS * 16);  // [N*2]

    Gate_29738353558213_gate_kernel<<<NUM_BLOCKS, 256, 0, stream>>>(
        x, wg, out_w, topE, meP, ceP);
    Gate_29738353558213_dispatch_kernel<<<NUM_EXPERTS, 256, 0, stream>>>(
        topE, out_idx, out_eid);
    Gate_29738353558213_laux_kernel<<<1, 32, 0, stream>>>(meP, ceP, out_laux);
}